// FourierFilter2D_17128329576597
// MI455X (gfx1250) — compile-verified
//
#include <hip/hip_runtime.h>

// Complex locally-connected channel mixing:
//   out[b,h,w,f] = sum_c x[b,h,w,c] * W[c,h,w,f] + bias[f]   (complex64)
// Per spatial position (h,w): complex GEMM  X[128,256] x W[256,256].
// fp32 WMMA (V_WMMA_F32_16X16X4_F32) keeps reference precision.
// v2 changes vs v1:
//  - flat grid with the 4 F-split blocks of a position adjacent -> A tile
//    (256 KB) is reused out of the 192 MB L2 instead of re-read from HBM.
//  - register double-buffering: global prefetch of tile k+KT overlaps the
//    512 WMMAs of tile k (v1 stalled on s_wait_loadcnt before every stage).

typedef __attribute__((ext_vector_type(2))) float v2f;
typedef __attribute__((ext_vector_type(4))) float v4f;
typedef __attribute__((ext_vector_type(8))) float v8f;

#define B_  128
#define H_  32
#define W_  32
#define C_  256
#define F_  256
#define XB_STRIDE  (H_*W_*C_)   // 262144: b-stride in x
#define WC_STRIDE  (H_*W_*F_)   // 262144: c-stride in w
#define OB_STRIDE  (H_*W_*F_)   // 262144: b-stride in out (complex elements)

#define KT  16                  // K tile (C chunk) staged in LDS
#define AST 20                  // padded LDS row stride (floats): 20*L mod 64 distinct, L=0..15
#define NTB 64                  // N (F) tile per block

__global__ __launch_bounds__(256)
void cplx_lc_gemm_wmma(const float* __restrict__ xr, const float* __restrict__ xi,
                       const float* __restrict__ wr, const float* __restrict__ wi,
                       const float* __restrict__ br, const float* __restrict__ bi,
                       float* __restrict__ out)
{
    __shared__ float sAr[B_ * AST];     // A tile [m][k], m=0..127, k=0..15
    __shared__ float sAi[B_ * AST];
    __shared__ float sBr[NTB * AST];    // B tile transposed [n][k], n=0..63
    __shared__ float sBi[NTB * AST];

    const int bid  = blockIdx.x;            // 0..4095, position-major
    const int p    = bid >> 2;              // h*W + w  (0..1023)
    const int n0   = (bid & 3) * NTB;       // F offset (0,64,128,192) - adjacent blocks
    const int tid  = threadIdx.x;
    const int wave = tid >> 5;
    const int lane = tid & 31;
    const int m0   = wave * 16;             // wave's M strip

    const int xbase = p * C_;               // x[., h, w, 0]
    const int wbase = p * F_ + n0;          // w[., h, w, n0]

    // staging-thread coordinates (fixed per thread)
    const int am0  = tid >> 2;              // A: m row for slot tid
    const int akq  = (tid & 3) << 2;        // A: k quad within tile
    const int am1  = (tid + 256) >> 2;      // A: m row for slot tid+256
    const int bcr  = tid >> 4;              // B: c row within tile (0..15)
    const int bfq  = (tid & 15) << 2;       // B: f quad within tile (0..60)

    // prefetch registers (double buffer in VGPRs)
    v4f pAr0, pAi0, pAr1, pAi1, pBr, pBi;

    #define LOAD_TILE(k0)                                                   \
        do {                                                                \
            const int g0 = xbase + am0 * XB_STRIDE + ((k0) + akq);          \
            const int g1 = xbase + am1 * XB_STRIDE + ((k0) + akq);          \
            pAr0 = *(const v4f*)(xr + g0);  pAi0 = *(const v4f*)(xi + g0);  \
            pAr1 = *(const v4f*)(xr + g1);  pAi1 = *(const v4f*)(xi + g1);  \
            const int gb = ((k0) + bcr) * WC_STRIDE + wbase + bfq;          \
            pBr = *(const v4f*)(wr + gb);   pBi = *(const v4f*)(wi + gb);   \
        } while (0)

    #define STORE_TILE()                                                    \
        do {                                                                \
            *(v4f*)&sAr[am0 * AST + akq] = pAr0;                            \
            *(v4f*)&sAi[am0 * AST + akq] = pAi0;                            \
            *(v4f*)&sAr[am1 * AST + akq] = pAr1;                            \
            *(v4f*)&sAi[am1 * AST + akq] = pAi1;                            \
            _Pragma("unroll")                                               \
            for (int j = 0; j < 4; ++j) {                                   \
                sBr[(bfq + j) * AST + bcr] = pBr[j];                        \
                sBi[(bfq + j) * AST + bcr] = pBi[j];                        \
            }                                                               \
        } while (0)

    v8f cr[4], ci[4];
    #pragma unroll
    for (int nt = 0; nt < 4; ++nt) { cr[nt] = (v8f){}; ci[nt] = (v8f){}; }

    LOAD_TILE(0);

    for (int k0 = 0; k0 < C_; k0 += KT) {
        STORE_TILE();
        __syncthreads();

        // kick off next tile's global loads; they complete during compute
        if (k0 + KT < C_) LOAD_TILE(k0 + KT);

        // ---- compute: 4 k-steps x 4 N-tiles x 4 WMMAs (complex MAC)
        const int mrow = m0 + (lane & 15);
        #pragma unroll
        for (int kk = 0; kk < KT; kk += 4) {
            // fp32 16x4 A layout: lanes 0-15 hold K=kk,kk+1; lanes 16-31 K=kk+2,kk+3
            const int ko = kk + ((lane >> 4) << 1);
            v2f ar = *(const v2f*)&sAr[mrow * AST + ko];
            v2f ai = *(const v2f*)&sAi[mrow * AST + ko];
            v2f an = -ai;                        // f32 WMMA NEG only applies to C
            #pragma unroll
            for (int nt = 0; nt < 4; ++nt) {
                const int nrow = nt * 16 + (lane & 15);
                v2f pr = *(const v2f*)&sBr[nrow * AST + ko];
                v2f pi = *(const v2f*)&sBi[nrow * AST + ko];
                // Cr += Ar*Br - Ai*Bi ; Ci += Ar*Bi + Ai*Br
                cr[nt] = __builtin_amdgcn_wmma_f32_16x16x4_f32(false, ar, false, pr, (short)0, cr[nt], false, false);
                cr[nt] = __builtin_amdgcn_wmma_f32_16x16x4_f32(false, an, false, pi, (short)0, cr[nt], false, false);
                ci[nt] = __builtin_amdgcn_wmma_f32_16x16x4_f32(false, ar, false, pi, (short)0, ci[nt], false, false);
                ci[nt] = __builtin_amdgcn_wmma_f32_16x16x4_f32(false, ai, false, pr, (short)0, ci[nt], false, false);
            }
        }
        __syncthreads();   // compute done before LDS is overwritten next iter
    }

    // ---- epilogue: C/D layout -> interleaved complex64, bias added.
    // VGPR r: lanes 0-15 => M=m0+r, lanes 16-31 => M=m0+8+r; N = lane&15.
    const int mo = m0 + ((lane >> 4) << 3);
    #pragma unroll
    for (int nt = 0; nt < 4; ++nt) {
        const int f   = n0 + nt * 16 + (lane & 15);
        const float fbr = br[f];
        const float fbi = bi[f];
        #pragma unroll
        for (int r = 0; r < 8; ++r) {
            const int oc = (mo + r) * OB_STRIDE + p * F_ + f;  // complex element index
            v2f v; v.x = cr[nt][r] + fbr; v.y = ci[nt][r] + fbi;
            *(v2f*)(out + 2 * (long)oc) = v;
        }
    }

    #undef LOAD_TILE
    #undef STORE_TILE
}

extern "C" void kernel_launch(void* const* d_in, const int* in_sizes, int n_in,
                              void* d_out, int out_size, void* d_ws, size_t ws_size,
                              hipStream_t stream) {
    const float* xr = (const float*)d_in[0];
    const float* xi = (const float*)d_in[1];
    const float* wr = (const float*)d_in[2];
    const float* wi = (const float*)d_in[3];
    const float* br = (const float*)d_in[4];
    const float* bi = (const float*)d_in[5];
    float* out = (float*)d_out;   // complex64 -> interleaved (re, im) float pairs

    dim3 grid(H_ * W_ * (F_ / NTB), 1, 1);   // 4096 blocks, position-major
    dim3 block(256, 1, 1);                   // 8 wave32s
    cplx_lc_gemm_wmma<<<grid, block, 0, stream>>>(xr, xi, wr, wi, br, bi, out);
}